// NumberPlateDetector_62921270886622
// MI455X (gfx1250) — compile-verified
//
#include <hip/hip_runtime.h>
#include <cstdint>

// ---------------------------------------------------------------------------
// Box decode (N x 5 AoS -> N x 4 xyxy + N keep flags), HBM-bandwidth bound
// (~189 MB total => ~8 us at 23.3 TB/s). No contraction dim => WMMA N/A.
// CDNA5 path: async DMA staging into LDS (GLOBAL_LOAD_ASYNC_TO_LDS_B128,
// ASYNCcnt), s_wait_asynccnt, 128-bit non-temporal stores. wave32-safe.
// 32-bit index math + uniform per-block full-tile fast path (no exec churn).
// ---------------------------------------------------------------------------

typedef float v4f __attribute__((ext_vector_type(4)));
typedef int   v4i __attribute__((ext_vector_type(4)));

#define THREADS           256
#define BOXES_PER_THREAD  4
#define BOXES_PER_BLOCK   (THREADS * BOXES_PER_THREAD)   // 1024 boxes / block
#define F4_PER_BLOCK      (BOXES_PER_BLOCK * 5 / 4)      // 1280 float4 = 20 KB

// Builtin signature (probed): non-const int4 AS1* src, int4 AS3* dst,
// imm offset, imm cpol.
typedef __attribute__((address_space(1))) v4i* gv4ip;
typedef __attribute__((address_space(3))) v4i* lv4ip;

__device__ __forceinline__ lv4ip as_lds_i4(const v4f* p) {
  // Generic LDS pointer's low 32 bits are the LDS byte offset (ISA 10.2).
  return (lv4ip)(uint32_t)(uintptr_t)p;
}
__device__ __forceinline__ gv4ip as_global_i4(const v4f* p) {
  return (gv4ip)(uintptr_t)p;
}

__device__ __forceinline__ void decode4(const v4f q0, const v4f q1, const v4f q2,
                                        const v4f q3, const v4f q4, float thr,
                                        v4f o[4], v4f* kv) {
  const float bx[4] = {q0.x, q1.y, q2.z, q3.w};
  const float by[4] = {q0.y, q1.z, q2.w, q4.x};
  const float bw[4] = {q0.z, q1.w, q3.x, q4.y};
  const float bh[4] = {q0.w, q2.x, q3.y, q4.z};
  const float bs[4] = {q1.x, q2.y, q3.z, q4.w};
#pragma unroll
  for (int b = 0; b < 4; ++b) {
    const bool keep = bs[b] > thr;
    const float hw = 0.5f * bw[b];
    const float hh = 0.5f * bh[b];
    o[b].x = keep ? (bx[b] - hw) : 0.0f;
    o[b].y = keep ? (by[b] - hh) : 0.0f;
    o[b].z = keep ? (bx[b] + hw) : 0.0f;
    o[b].w = keep ? (by[b] + hh) : 0.0f;
    ((float*)kv)[b] = keep ? 1.0f : 0.0f;
  }
}

__global__ __launch_bounds__(THREADS)
void box_decode_kernel(const float* __restrict__ X,
                       const float* __restrict__ thrp,
                       float* __restrict__ out,
                       int nBoxes) {
  __shared__ v4f tile[F4_PER_BLOCK];

  const int t = threadIdx.x;                      // 0..255
  const int blockBoxBase = (int)blockIdx.x * BOXES_PER_BLOCK;
  const int blockF4Base  = blockBoxBase / 4 * 5;  // float4 index into input
  const int totalF4      = nBoxes / 4 * 5;
  const v4f* in4 = (const v4f*)X;

  v4f* out4  = (v4f*)out;                         // xyxy: nBoxes * float4
  float* keepBase = out + (size_t)4 * (unsigned)nBoxes;
  v4f* keep4 = (v4f*)keepBase;

  const float thr = thrp[0];

  if (blockBoxBase + BOXES_PER_BLOCK <= nBoxes) {
    // ---- fast path: full tile, uniform (scalar) branch, no exec masking ----
    // Async stage: 1280 contiguous float4 / block; 512B per wave-issue.
#pragma unroll
    for (int k = 0; k < 5; ++k) {
      __builtin_amdgcn_global_load_async_to_lds_b128(
          as_global_i4(in4 + (blockF4Base + k * THREADS + t)),
          as_lds_i4(&tile[k * THREADS + t]), 0, 0);
    }
    asm volatile("s_wait_asynccnt 0x0" ::: "memory");
    __syncthreads();

    // Thread t owns boxes [4t,4t+4): 20 floats = 5 float4 in LDS (AoS repack).
    v4f o[4], kv;
    decode4(tile[t * 5 + 0], tile[t * 5 + 1], tile[t * 5 + 2],
            tile[t * 5 + 3], tile[t * 5 + 4], thr, o, &kv);

    const int boxBase = blockBoxBase + t * 4;
#pragma unroll
    for (int b = 0; b < 4; ++b) {
      __builtin_nontemporal_store(o[b], out4 + (boxBase + b)); // b128 NT
    }
    __builtin_nontemporal_store(kv, keep4 + (blockBoxBase / 4 + t));
  } else {
    // ---- tail path (never taken for N % 1024 == 0): plain guarded loads ----
    const int boxBase = blockBoxBase + t * 4;
#pragma unroll
    for (int b = 0; b < 4; ++b) {
      const int box = boxBase + b;
      if (box < nBoxes) {
        const float x = X[(size_t)box * 5 + 0];
        const float y = X[(size_t)box * 5 + 1];
        const float w = X[(size_t)box * 5 + 2];
        const float h = X[(size_t)box * 5 + 3];
        const float s = X[(size_t)box * 5 + 4];
        const bool keep = s > thr;
        v4f o;
        o.x = keep ? (x - 0.5f * w) : 0.0f;
        o.y = keep ? (y - 0.5f * h) : 0.0f;
        o.z = keep ? (x + 0.5f * w) : 0.0f;
        o.w = keep ? (y + 0.5f * h) : 0.0f;
        __builtin_nontemporal_store(o, out4 + box);
        keepBase[box] = keep ? 1.0f : 0.0f;
      }
    }
  }
}

extern "C" void kernel_launch(void* const* d_in, const int* in_sizes, int n_in,
                              void* d_out, int out_size, void* d_ws, size_t ws_size,
                              hipStream_t stream) {
  (void)n_in; (void)out_size; (void)d_ws; (void)ws_size;
  const float* X   = (const float*)d_in[0];
  const float* thr = (const float*)d_in[1];
  float* out = (float*)d_out;

  const int nBoxes = in_sizes[0] / 5;                            // 4,718,592
  const int blocks = (nBoxes + BOXES_PER_BLOCK - 1) / BOXES_PER_BLOCK;

  hipLaunchKernelGGL(box_decode_kernel, dim3(blocks), dim3(THREADS), 0, stream,
                     X, thr, out, nBoxes);
}